// LightGCN_46325517254676
// MI455X (gfx1250) — compile-verified
//
#include <hip/hip_runtime.h>

#define N_USER  200000
#define N_ITEM  100000
#define N_TOTAL (N_USER + N_ITEM)
#define DIM     64
#define N_LAYER 3
#define N_EDGES 9600000
#define BATCH   16384

typedef __attribute__((ext_vector_type(2))) float v2f;
typedef __attribute__((ext_vector_type(4))) float v4f;
typedef __attribute__((ext_vector_type(8))) float v8f;

// x = acc = concat(emb_u, emb_i); y = 0.  emb read once -> non-temporal.
__global__ void lg_init(const float* __restrict__ emb_u,
                        const float* __restrict__ emb_i,
                        float* __restrict__ x,
                        float* __restrict__ acc,
                        float* __restrict__ y) {
  size_t i = (size_t)blockIdx.x * blockDim.x + threadIdx.x;   // float4 index
  const size_t nd4 = (size_t)N_TOTAL * DIM / 4;
  if (i >= nd4) return;
  const size_t u4 = (size_t)N_USER * DIM / 4;
  v4f v = (i < u4) ? __builtin_nontemporal_load((const v4f*)emb_u + i)
                   : __builtin_nontemporal_load((const v4f*)emb_i + (i - u4));
  ((v4f*)x)[i]   = v;
  ((v4f*)acc)[i] = v;
  ((v4f*)y)[i]   = (v4f){0.f, 0.f, 0.f, 0.f};
}

__global__ void lg_mask_zero(unsigned char* __restrict__ mask) {
  int i = blockIdx.x * blockDim.x + threadIdx.x;
  if (i < N_TOTAL) mask[i] = 0;
}

// Mark rows needed by the final gather (benign same-value write races).
__global__ void lg_mask_mark(const int* __restrict__ idx_u,
                             const int* __restrict__ idx_i,
                             unsigned char* __restrict__ mask) {
  int b = blockIdx.x * blockDim.x + threadIdx.x;
  if (b >= BATCH) return;
  mask[idx_u[b]] = 1;
  mask[N_USER + idx_i[b]] = 1;
}

// Dense edge-parallel SpMM: 16 threads/edge, one float4 each (256B/row fully
// coalesced). Edge lists are streamed non-temporally so the 192MB L2 holds
// the feature matrices (gather src + atomic dst) instead.
__global__ void lg_spmm(const int* __restrict__ rows,
                        const int* __restrict__ cols,
                        const float* __restrict__ vals,
                        const float* __restrict__ x,
                        float* __restrict__ y) {
  size_t t = (size_t)blockIdx.x * blockDim.x + threadIdx.x;
  size_t e = t >> 4;
  int seg  = (int)(t & 15);
  if (e >= (size_t)N_EDGES) return;
  int   r = __builtin_nontemporal_load(rows + e);
  int   c = __builtin_nontemporal_load(cols + e);
  float v = __builtin_nontemporal_load(vals + e);
  v4f xv = ((const v4f*)(x + (size_t)c * DIM))[seg];
  float* yr = y + (size_t)r * DIM + (size_t)seg * 4;
  atomicAdd(yr + 0, v * xv.x);
  atomicAdd(yr + 1, v * xv.y);
  atomicAdd(yr + 2, v * xv.z);
  atomicAdd(yr + 3, v * xv.w);
}

// Last layer: only rows consumed by the final gather matter (~11% of rows),
// so skip edges whose destination row is unmasked (~79% of waves drop all
// feature traffic).
__global__ void lg_spmm_masked(const int* __restrict__ rows,
                               const int* __restrict__ cols,
                               const float* __restrict__ vals,
                               const float* __restrict__ x,
                               float* __restrict__ y,
                               const unsigned char* __restrict__ mask) {
  size_t t = (size_t)blockIdx.x * blockDim.x + threadIdx.x;
  size_t e = t >> 4;
  int seg  = (int)(t & 15);
  if (e >= (size_t)N_EDGES) return;
  int r = __builtin_nontemporal_load(rows + e);
  if (!mask[r]) return;
  int   c = __builtin_nontemporal_load(cols + e);
  float v = __builtin_nontemporal_load(vals + e);
  v4f xv = ((const v4f*)(x + (size_t)c * DIM))[seg];
  float* yr = y + (size_t)r * DIM + (size_t)seg * 4;
  atomicAdd(yr + 0, v * xv.x);
  atomicAdd(yr + 1, v * xv.y);
  atomicAdd(yr + 2, v * xv.z);
  atomicAdd(yr + 3, v * xv.w);
}

// acc += y, and zero the next layer's target buffer in the same pass.
__global__ void lg_add_zero(float* __restrict__ acc,
                            const float* __restrict__ y,
                            float* __restrict__ xz) {
  size_t i = (size_t)blockIdx.x * blockDim.x + threadIdx.x;
  const size_t nd4 = (size_t)N_TOTAL * DIM / 4;
  if (i >= nd4) return;
  v4f a = ((const v4f*)acc)[i];
  v4f b = ((const v4f*)y)[i];
  a = a + b;
  ((v4f*)acc)[i] = a;
  ((v4f*)xz)[i]  = (v4f){0.f, 0.f, 0.f, 0.f};
}

// Batched dot on the matrix pipe: per wave32, 16 batch elements.
// Feature row = acc(x0+x1+x2) + y3 (layer-3 output, folded here so the dense
// layer-3 acc pass is skipped).  A[M,K]=row_u[M][K], B[K,N]=row_i[N][K];
// 16 chained V_WMMA_F32_16X16X4_F32 give all 16x16 dots; keep the diagonal,
// scaled by 1/16 (= (1/(N_LAYER+1))^2, one 1/4 per factor).
//
// 16x4 f32 A / 4x16 f32 B layouts both reduce to: lane L loads float2 at
// column 4*kk + 2*(L>>4) of its row (row index L&15) -> no LDS staging.
__global__ void __launch_bounds__(256)
lg_dot_wmma(const float* __restrict__ acc,
            const float* __restrict__ y3,
            const int* __restrict__ idx_u,
            const int* __restrict__ idx_i,
            float* __restrict__ out) {
  int wave = (int)((blockIdx.x * blockDim.x + threadIdx.x) >> 5); // tile id
  int lane = (int)(threadIdx.x & 31);
  int m    = lane & 15;
  int half = lane >> 4;
  int base = wave * 16;

  size_t urow = (size_t)idx_u[base + m] * DIM;
  size_t irow = ((size_t)N_USER + (size_t)idx_i[base + m]) * DIM;

  const float* au = acc + urow;
  const float* yu = y3  + urow;
  const float* bi = acc + irow;
  const float* yi = y3  + irow;

  v8f c = {};
#pragma unroll
  for (int kk = 0; kk < 16; ++kk) {
    int k0 = kk * 4 + half * 2;
    v2f a = *(const v2f*)(au + k0) + *(const v2f*)(yu + k0);
    v2f b = *(const v2f*)(bi + k0) + *(const v2f*)(yi + k0);
    // (neg_a, A, neg_b, B, c_mod, C, reuse_a, reuse_b)
    c = __builtin_amdgcn_wmma_f32_16x16x4_f32(false, a, false, b,
                                              (short)0, c, false, false);
  }

  // Diagonal of 16x16 f32 D: VGPR v, lanes 0-15 -> M=v,N=lane; lanes 16-31
  // -> M=v+8,N=lane-16.  M==N at (v=lane, lane<8) and (v=lane-24, lane>=24).
  const float scale = 1.0f / 16.0f;
  if (lane < 8 || lane >= 24) {
    int sel = (lane < 16) ? lane : (lane - 24);
    float d = c[sel & 7];
    out[base + (lane & 15)] = d * scale;
  }
}

extern "C" void kernel_launch(void* const* d_in, const int* in_sizes, int n_in,
                              void* d_out, int out_size, void* d_ws, size_t ws_size,
                              hipStream_t stream) {
  const float* emb_u = (const float*)d_in[0];
  const float* emb_i = (const float*)d_in[1];
  const int*   rows  = (const int*)d_in[2];
  const int*   cols  = (const int*)d_in[3];
  const float* vals  = (const float*)d_in[4];
  const int*   idx_u = (const int*)d_in[5];
  const int*   idx_i = (const int*)d_in[6];
  float*       out   = (float*)d_out;

  const size_t nd = (size_t)N_TOTAL * DIM;   // 19.2M floats per buffer
  float* bufA = (float*)d_ws;
  float* bufB = bufA + nd;
  float* acc  = bufB + nd;
  unsigned char* mask = (unsigned char*)(acc + nd);

  const int blk = 256;
  const size_t nd4 = nd / 4;
  dim3 gridElem((unsigned)((nd4 + blk - 1) / blk));
  const size_t spmm_threads = (size_t)N_EDGES * 16;
  dim3 gridSpmm((unsigned)((spmm_threads + blk - 1) / blk));

  // x(bufA) = acc = emb; zero layer-1 target (bufB); build row mask.
  lg_init<<<gridElem, blk, 0, stream>>>(emb_u, emb_i, bufA, acc, bufB);
  lg_mask_zero<<<(N_TOTAL + blk - 1) / blk, blk, 0, stream>>>(mask);
  lg_mask_mark<<<(BATCH + blk - 1) / blk, blk, 0, stream>>>(idx_u, idx_i, mask);

  // Layer 1: x=bufA -> y=bufB; acc += bufB; zero bufA (layer-2 target).
  lg_spmm<<<gridSpmm, blk, 0, stream>>>(rows, cols, vals, bufA, bufB);
  lg_add_zero<<<gridElem, blk, 0, stream>>>(acc, bufB, bufA);

  // Layer 2: x=bufB -> y=bufA; acc += bufA; zero bufB (layer-3 target).
  lg_spmm<<<gridSpmm, blk, 0, stream>>>(rows, cols, vals, bufB, bufA);
  lg_add_zero<<<gridElem, blk, 0, stream>>>(acc, bufA, bufB);

  // Layer 3: only rows feeding the final gather; y3 stays in bufB and is
  // folded into the dot kernel (no dense acc pass).
  lg_spmm_masked<<<gridSpmm, blk, 0, stream>>>(rows, cols, vals, bufA, bufB, mask);

  // 1024 tiles of 16 batch elements, 8 waves (256 threads) per block.
  lg_dot_wmma<<<BATCH / 16 / 8, 256, 0, stream>>>(acc, bufB, idx_u, idx_i, out);
}